// CondConv_27994596836025
// MI455X (gfx1250) — compile-verified
//
#include <hip/hip_runtime.h>
#include <hip/hip_bf16.h>
#include <math.h>

// ---------------------------------------------------------------------------
// CondConv on gfx1250 (MI455X): implicit-GEMM conv using V_WMMA_F32_16X16X4_F32
// ---------------------------------------------------------------------------

typedef __attribute__((ext_vector_type(2))) float v2f;
typedef __attribute__((ext_vector_type(8))) float v8f;

#define BATCH 16
#define CIN   256
#define COUT  256
#define HH    64
#define WW    64
#define KK    3
#define NE    3

// conv tiling
#define BM       128   // C_out per block
#define ROWS_PB  4     // output rows per block  -> BN = 4*64 = 256
#define CH       32    // channel chunk staged in LDS
#define CHP      34    // padded ci stride (even -> b64-aligned; 34 mod 64 walk
                       //  covers 32 distinct banks over 32 consecutive rows)
#define A_STRIDE 34
#define XROW     66    // 64 cols + 1 halo each side
#define XSLICE   (ROWS_PB * XROW)   // 264 (row,col) positions per chunk

// ---------------------------------------------------------------------------
// Kernel 1: global average pool  pooled[b][c] = mean_{h,w} x[b][c][h][w]
// ---------------------------------------------------------------------------
__global__ __launch_bounds__(256)
void cc_pool_kernel(const float* __restrict__ x, float* __restrict__ pooled) {
    const int bc  = blockIdx.x;               // b*CIN + c
    const float* p = x + (size_t)bc * (HH * WW);
    float s = 0.f;
    for (int i = threadIdx.x; i < HH * WW; i += 256) s += p[i];
    for (int off = 16; off > 0; off >>= 1) s += __shfl_down(s, off, 32);
    __shared__ float red[8];
    const int lane = threadIdx.x & 31, wave = threadIdx.x >> 5;
    if (lane == 0) red[wave] = s;
    __syncthreads();
    if (threadIdx.x == 0) {
        float t = 0.f;
        #pragma unroll
        for (int i = 0; i < 8; ++i) t += red[i];
        pooled[bc] = t * (1.0f / (HH * WW));
    }
}

// ---------------------------------------------------------------------------
// Kernel 2: routing weights  rw[b][e] = sigmoid(pooled[b]·fc_w[e] + fc_b[e])
// ---------------------------------------------------------------------------
__global__ __launch_bounds__(256)
void cc_route_kernel(const float* __restrict__ pooled,
                     const float* __restrict__ fc_w,
                     const float* __restrict__ fc_b,
                     float* __restrict__ rw) {
    const int be = blockIdx.x;                // b*NE + e
    const int b = be / NE, e = be % NE;
    float s = pooled[b * CIN + threadIdx.x] * fc_w[e * CIN + threadIdx.x];
    for (int off = 16; off > 0; off >>= 1) s += __shfl_down(s, off, 32);
    __shared__ float red[8];
    const int lane = threadIdx.x & 31, wave = threadIdx.x >> 5;
    if (lane == 0) red[wave] = s;
    __syncthreads();
    if (threadIdx.x == 0) {
        float t = 0.f;
        #pragma unroll
        for (int i = 0; i < 8; ++i) t += red[i];
        t += fc_b[e];
        rw[be] = 1.0f / (1.0f + __expf(-t));
    }
}

// ---------------------------------------------------------------------------
// Kernel 3: implicit-GEMM conv + BN + SiLU, f32 WMMA 16x16x4.
// grid = (H/ROWS_PB, COUT/BM, B); block = 256 threads = 8 waves.
// Each wave computes a 32(M) x 128(N) output region = 2x8 WMMA tiles.
// ---------------------------------------------------------------------------
__global__ __launch_bounds__(256)
void cc_conv_wmma_kernel(const float* __restrict__ x,
                         const float* __restrict__ experts,
                         const float* __restrict__ rw,
                         const float* __restrict__ bn_gamma,
                         const float* __restrict__ bn_beta,
                         const float* __restrict__ bn_mean,
                         const float* __restrict__ bn_var,
                         float* __restrict__ out) {
    __shared__ __align__(16) float a_s[KK * BM * A_STRIDE];   // [kw][o][ci]
    __shared__ __align__(16) float x_s[XSLICE * CHP];         // [(row,col)][ci]
    __shared__ float inv_s[BM], add_s[BM];

    const int tid = threadIdx.x;
    const int hb  = blockIdx.x;        // 0..15
    const int ob  = blockIdx.y;        // 0..1
    const int b   = blockIdx.z;        // 0..15
    const int o0  = ob * BM;
    const int h0  = hb * ROWS_PB;

    const float r0 = rw[b * NE + 0];
    const float r1 = rw[b * NE + 1];
    const float r2 = rw[b * NE + 2];

    // fused BN epilogue constants for this block's 128 output channels
    for (int i = tid; i < BM; i += 256) {
        const float inv = bn_gamma[o0 + i] * rsqrtf(bn_var[o0 + i] + 1e-5f);
        inv_s[i] = inv;
        add_s[i] = bn_beta[o0 + i] - bn_mean[o0 + i] * inv;
    }

    const int wave  = tid >> 5;
    const int lane  = tid & 31;
    const int wm    = wave & 3;        // M offset = 32*wm
    const int wn    = wave >> 2;       // N offset = 128*wn
    const int m_l   = lane & 15;       // row/col within 16-tile
    const int hhalf = lane >> 4;       // lane half selects K pair / M half

    v8f acc[2][8];
    #pragma unroll
    for (int mt = 0; mt < 2; ++mt)
        #pragma unroll
        for (int nt = 0; nt < 8; ++nt)
            acc[mt][nt] = (v8f)0.0f;

    const int estr = COUT * CIN * KK * KK;

    for (int kh = 0; kh < KK; ++kh) {
        for (int ci0 = 0; ci0 < CIN; ci0 += CH) {
            __syncthreads();
            // ---- stage X chunk (transposed): x_s[(rl*66+wi)][ci] ----
            for (int idx = tid; idx < CH * XSLICE; idx += 256) {
                const int ci  = idx / XSLICE;
                const int rem = idx - ci * XSLICE;
                const int rl  = rem / XROW;
                const int wi  = rem - rl * XROW;
                const int row = h0 + kh - 1 + rl;
                const int col = wi - 1;
                float v = 0.f;
                if ((unsigned)row < (unsigned)HH && (unsigned)col < (unsigned)WW)
                    v = x[(((b * CIN) + ci0 + ci) * HH + row) * WW + col];
                x_s[rem * CHP + ci] = v;
            }
            // ---- stage combined weights: sum_e rw[b,e]*experts[e][o][ci][kh][kw] ----
            for (int idx = tid; idx < KK * BM * CH; idx += 256) {
                const int kw  = idx / (BM * CH);
                const int rem = idx - kw * (BM * CH);
                const int o   = rem / CH;
                const int ci  = rem - o * CH;
                const int base = (((o0 + o) * CIN) + ci0 + ci) * (KK * KK) + kh * KK + kw;
                const float v = r0 * experts[base]
                              + r1 * experts[base + estr]
                              + r2 * experts[base + 2 * estr];
                a_s[(kw * BM + o) * A_STRIDE + ci] = v;
            }
            // ---- prefetch next chunk toward L2/L0 (global_prefetch_b8) ----
            {
                const int nci0 = ci0 + CH;
                if (nci0 < CIN) {
                    for (int t = tid; t < BM; t += 256)
                        __builtin_prefetch(
                            &experts[(((o0 + t) * CIN) + nci0) * (KK * KK) + kh * KK], 0, 1);
                    for (int t = tid; t < CH * ROWS_PB; t += 256) {
                        const int ci = t >> 2, rl = t & 3;
                        const int row = h0 + kh - 1 + rl;
                        if ((unsigned)row < (unsigned)HH)
                            __builtin_prefetch(
                                &x[(((b * CIN) + nci0 + ci) * HH + row) * WW], 0, 1);
                    }
                }
            }
            __syncthreads();

            for (int kw = 0; kw < KK; ++kw) {
                const float* ab = &a_s[kw * BM * A_STRIDE];
                for (int k4 = 0; k4 < CH; k4 += 4) {
                    const int kv = k4 + 2 * hhalf;            // even -> 8B aligned
                    // A fragments (16x4): a[v] = A[m][kv+v]   (ds_load_b64)
                    v2f afr[2];
                    afr[0] = *(const v2f*)&ab[(32 * wm +      m_l) * A_STRIDE + kv];
                    afr[1] = *(const v2f*)&ab[(32 * wm + 16 + m_l) * A_STRIDE + kv];
                    // B fragments (4x16): b[v] = B[kv+v][n]   (ds_load_b64, transposed x_s)
                    v2f bfr[8];
                    #pragma unroll
                    for (int nt = 0; nt < 8; ++nt) {
                        const int n  = 128 * wn + 16 * nt + m_l;
                        const int rl = n >> 6;
                        const int w  = n & 63;
                        bfr[nt] = *(const v2f*)&x_s[(rl * XROW + w + kw) * CHP + kv];
                    }
                    #pragma unroll
                    for (int mt = 0; mt < 2; ++mt)
                        #pragma unroll
                        for (int nt = 0; nt < 8; ++nt)
                            acc[mt][nt] = __builtin_amdgcn_wmma_f32_16x16x4_f32(
                                false, afr[mt], false, bfr[nt],
                                (short)0, acc[mt][nt], false, false);
                }
            }
        }
    }

    // ---- epilogue: BN (eval) + SiLU, write out[b][o][h][w] ----
    #pragma unroll
    for (int mt = 0; mt < 2; ++mt) {
        #pragma unroll
        for (int nt = 0; nt < 8; ++nt) {
            const int n  = 128 * wn + 16 * nt + m_l;
            const int rl = n >> 6;
            const int w  = n & 63;
            #pragma unroll
            for (int v = 0; v < 8; ++v) {
                const int ol = 32 * wm + 16 * mt + 8 * hhalf + v;  // 0..127
                float y = acc[mt][nt][v];
                y = y * inv_s[ol] + add_s[ol];
                y = y / (1.0f + __expf(-y));                       // y*sigmoid(y)
                out[(((b * COUT) + o0 + ol) * HH + (h0 + rl)) * WW + w] = y;
            }
        }
    }
}

// ---------------------------------------------------------------------------
// Launch
// ---------------------------------------------------------------------------
extern "C" void kernel_launch(void* const* d_in, const int* in_sizes, int n_in,
                              void* d_out, int out_size, void* d_ws, size_t ws_size,
                              hipStream_t stream) {
    const float* x        = (const float*)d_in[0];
    const float* experts  = (const float*)d_in[1];
    const float* fc_w     = (const float*)d_in[2];
    const float* fc_b     = (const float*)d_in[3];
    const float* bn_gamma = (const float*)d_in[4];
    const float* bn_beta  = (const float*)d_in[5];
    const float* bn_mean  = (const float*)d_in[6];
    const float* bn_var   = (const float*)d_in[7];
    float* out = (float*)d_out;

    float* pooled = (float*)d_ws;              // B*CIN = 4096 floats
    float* rw     = pooled + BATCH * CIN;      // B*NE  = 48 floats

    cc_pool_kernel<<<BATCH * CIN, 256, 0, stream>>>(x, pooled);
    cc_route_kernel<<<BATCH * NE, 256, 0, stream>>>(pooled, fc_w, fc_b, rw);

    dim3 grid(HH / ROWS_PB, COUT / BM, BATCH);  // (16, 2, 16)
    cc_conv_wmma_kernel<<<grid, 256, 0, stream>>>(
        x, experts, rw, bn_gamma, bn_beta, bn_mean, bn_var, out);
}